// old_VectorQuantizer_34402688041325
// MI455X (gfx1250) — compile-verified
//
#include <hip/hip_runtime.h>
#include <hip/hip_bf16.h>
#include <math.h>

typedef __attribute__((ext_vector_type(2))) float v2f;
typedef __attribute__((ext_vector_type(8))) float v8f;

#define N_TOK   32768
#define DIMS    256
#define KCODES  4096
#define HW      4096      // 64*64
#define ROWS_PER_WG 128   // 4 waves x 32 rows
#define BCODES  32        // codes per k-iteration (2 sub-tiles of 16)
#define NKT     (KCODES / BCODES)
#define APAD    260       // padded LDS row stride (floats): conflict-free b64 reads

// workspace layout (bytes)
#define WS_IDX      0                          // int[32768]
#define WS_COUNTS   131072                     // uint[4096]
#define WS_PARTIALS (131072 + 16384)           // float[8192]
#define WS_NORMS    (131072 + 16384 + 32768)   // float[4096]

// output offsets (in floats, concatenated return order)
#define OFF_LOSS  0
#define OFF_QUANT 1
#define OFF_PERP  8388609
#define OFF_ENC   ((size_t)8388610)
#define OFF_IDX   ((size_t)142606338)

__global__ void vq_zero_counts(unsigned int* __restrict__ counts) {
    int i = blockIdx.x * blockDim.x + threadIdx.x;
    if (i < KCODES) counts[i] = 0u;
}

__global__ void vq_cb_norms(const float* __restrict__ cb, float* __restrict__ norms) {
    int k = blockIdx.x * blockDim.x + threadIdx.x;
    if (k >= KCODES) return;
    const float* row = cb + (size_t)k * DIMS;
    float s = 0.f;
#pragma unroll 8
    for (int d = 0; d < DIMS; ++d) { float v = row[d]; s += v * v; }
    norms[k] = s;
}

// LDS offset of a generic pointer derived from LDS: aperture tag lives in addr[63:32],
// so the low 32 bits are the raw LDS byte offset.
__device__ __forceinline__ unsigned lds_off(const void* p) {
    return (unsigned)(uintptr_t)p;
}

// Async copy of one 16-byte chunk from global to LDS (tracked by ASYNCcnt).
__device__ __forceinline__ void async_copy_b128(const float4* g, float* lds_dst) {
    unsigned l = lds_off(lds_dst);
    asm volatile("global_load_async_to_lds_b128 %0, %1, off"
                 :: "v"(l), "v"(g) : "memory");
}

__device__ __forceinline__ void wait_async0() {
    asm volatile("s_wait_asynccnt 0x0" ::: "memory");
}

// Distance argmin via fp32 WMMA: score(n,k) = ||e_k||^2 - 2 * x_n . e_k
// 4 waves; each wave owns 32 rows and computes a 32x32 block per k-iteration
// (2x2 register blocking -> 1 LDS b64 per WMMA). B tile double-buffered via
// async-to-LDS loads overlapped with the WMMA loop.
__global__ __launch_bounds__(128) void vq_argmin_kernel(
    const float* __restrict__ z, const float* __restrict__ cb,
    const float* __restrict__ norms, int* __restrict__ idx_ws,
    unsigned int* __restrict__ counts, float* __restrict__ idx_out)
{
    extern __shared__ float smem[];
    float* ALDS  = smem;                              // 128 rows x APAD
    float* BLDS0 = smem + ROWS_PER_WG * APAD;         // 32 rows x APAD
    float* BLDS1 = BLDS0 + BCODES * APAD;             // 32 rows x APAD

    const int t = threadIdx.x;
    const int wgRow0 = blockIdx.x * ROWS_PER_WG;

    // Kick off async staging of B tile 0 (overlaps with synchronous A staging).
    {
        const float4* src = (const float4*)cb;        // kbase = 0
#pragma unroll
        for (int u = 0; u < 16; ++u) {
            int i = t + 128 * u;                      // float4 index 0..2047
            int row = i >> 6;
            int d4 = (i & 63) << 2;
            async_copy_b128(src + i, BLDS0 + row * APAD + d4);
        }
    }

    // Stage A: 128 rows of "flattened transposed" z. x[n][d] = z[b*D*HW + d*HW + hw]
    {
        int r = t;                                    // one row per thread
        int n = wgRow0 + r;
        int b = n >> 12;
        int hw = n & (HW - 1);
        const float* src = z + (size_t)b * DIMS * HW + hw;
#pragma unroll 4
        for (int d = 0; d < DIMS; ++d)
            ALDS[r * APAD + d] = src[(size_t)d * HW]; // coalesced across threads per d
    }

    wait_async0();
    __syncthreads();

    const int lane = t & 31;
    const int wave = t >> 5;
    const int lm = lane & 15;     // row/code within 16-group
    const int kh = lane >> 4;     // K-half select (A/B frag: K = v + 2*(lane/16))
    const float* aPtr0 = ALDS + (wave * 32 + lm) * APAD + 2 * kh;
    const float* aPtr1 = aPtr0 + 16 * APAD;
    const int bOff = lm * APAD + 2 * kh;

    float best0[8], best1[8];
    int   bk0[8],   bk1[8];
#pragma unroll
    for (int j = 0; j < 8; ++j) {
        best0[j] = 3.4e38f; best1[j] = 3.4e38f; bk0[j] = 0; bk1[j] = 0;
    }

    for (int kt = 0; kt < NKT; ++kt) {
        const float* Bcur = (kt & 1) ? BLDS1 : BLDS0;
        float*       Bnxt = (kt & 1) ? BLDS0 : BLDS1;

        // Prefetch next B tile asynchronously while we compute on the current one.
        if (kt + 1 < NKT) {
            const float4* src = (const float4*)(cb + (size_t)(kt + 1) * BCODES * DIMS);
#pragma unroll
            for (int u = 0; u < 16; ++u) {
                int i = t + 128 * u;
                int row = i >> 6;
                int d4 = (i & 63) << 2;
                async_copy_b128(src + i, Bnxt + row * APAD + d4);
            }
        }

        const float* bPtr0 = Bcur + bOff;
        const float* bPtr1 = bPtr0 + 16 * APAD;

        v8f acc00 = {}, acc01 = {}, acc10 = {}, acc11 = {};
#pragma unroll 16
        for (int s = 0; s < 64; ++s) {
            v2f a0 = *(const v2f*)(aPtr0 + 4 * s);
            v2f a1 = *(const v2f*)(aPtr1 + 4 * s);
            v2f b0 = *(const v2f*)(bPtr0 + 4 * s);
            v2f b1 = *(const v2f*)(bPtr1 + 4 * s);
            acc00 = __builtin_amdgcn_wmma_f32_16x16x4_f32(false, a0, false, b0, (short)0, acc00, false, false);
            acc01 = __builtin_amdgcn_wmma_f32_16x16x4_f32(false, a0, false, b1, (short)0, acc01, false, false);
            acc10 = __builtin_amdgcn_wmma_f32_16x16x4_f32(false, a1, false, b0, (short)0, acc10, false, false);
            acc11 = __builtin_amdgcn_wmma_f32_16x16x4_f32(false, a1, false, b1, (short)0, acc11, false, false);
        }

        const int kbase = kt * BCODES;
        const float nk0 = norms[kbase + lm];
        const float nk1 = norms[kbase + 16 + lm];
        const int k0 = kbase + lm;
        const int k1 = kbase + 16 + lm;
#pragma unroll
        for (int j = 0; j < 8; ++j) {
            // band 0 (rows wave*32 + [0,16))
            float sA = nk0 - 2.0f * acc00[j]; int kA = k0;
            float sB = nk1 - 2.0f * acc01[j];
            if (sB < sA) { sA = sB; kA = k1; }         // k0 < k1: strict keeps first-min
            if (sA < best0[j]) { best0[j] = sA; bk0[j] = kA; }
            // band 1 (rows wave*32 + [16,32))
            float sC = nk0 - 2.0f * acc10[j]; int kC = k0;
            float sD = nk1 - 2.0f * acc11[j];
            if (sD < sC) { sC = sD; kC = k1; }
            if (sC < best1[j]) { best1[j] = sC; bk1[j] = kC; }
        }

        wait_async0();      // own async loads into Bnxt done
        __syncthreads();    // all waves done computing on Bcur and loading Bnxt
    }

    // butterfly min+argmin across the 16 lanes of each half-wave group
#pragma unroll
    for (int j = 0; j < 8; ++j) {
        float s0 = best0[j]; int K0 = bk0[j];
        float s1 = best1[j]; int K1 = bk1[j];
#pragma unroll
        for (int m = 8; m >= 1; m >>= 1) {
            float so0 = __shfl_xor(s0, m, 32); int ko0 = __shfl_xor(K0, m, 32);
            if (so0 < s0 || (so0 == s0 && ko0 < K0)) { s0 = so0; K0 = ko0; }
            float so1 = __shfl_xor(s1, m, 32); int ko1 = __shfl_xor(K1, m, 32);
            if (so1 < s1 || (so1 == s1 && ko1 < K1)) { s1 = so1; K1 = ko1; }
        }
        bk0[j] = K0; bk1[j] = K1;
    }

    if (lane == 0 || lane == 16) {
        // C layout: VGPR j, half kh -> matrix row m = j + 8*kh
#pragma unroll
        for (int j = 0; j < 8; ++j) {
            int n0 = wgRow0 + wave * 32 + kh * 8 + j;      // band 0
            int k0w = bk0[j];
            idx_ws[n0] = k0w;
            idx_out[n0] = (float)k0w;
            atomicAdd(&counts[k0w], 1u);
            int n1 = n0 + 16;                              // band 1
            int k1w = bk1[j];
            idx_ws[n1] = k1w;
            idx_out[n1] = (float)k1w;
            atomicAdd(&counts[k1w], 1u);
        }
    }
}

// Gather codebook rows -> quantized [B,C,H,W]; deterministic SSE partials per block
__global__ __launch_bounds__(256) void vq_quantize_loss(
    const float* __restrict__ z, const float* __restrict__ cb,
    const int* __restrict__ idx_ws, float* __restrict__ quant_out,
    float* __restrict__ partials)
{
    __shared__ float red[256];
    const int t = threadIdx.x;
    const size_t e = ((size_t)blockIdx.x * 256 + t) * 4;  // element index in [B,C,H,W]
    const int hw = (int)(e & (HW - 1));
    const int c  = (int)((e >> 12) & 255);
    const int b  = (int)(e >> 20);
    const int n0 = (b << 12) + hw;

    float4 zv = *(const float4*)(z + e);
    float q0 = cb[(size_t)idx_ws[n0 + 0] * DIMS + c];
    float q1 = cb[(size_t)idx_ws[n0 + 1] * DIMS + c];
    float q2 = cb[(size_t)idx_ws[n0 + 2] * DIMS + c];
    float q3 = cb[(size_t)idx_ws[n0 + 3] * DIMS + c];
    quant_out[e + 0] = q0;
    quant_out[e + 1] = q1;
    quant_out[e + 2] = q2;
    quant_out[e + 3] = q3;
    float d0 = q0 - zv.x, d1 = q1 - zv.y, d2 = q2 - zv.z, d3 = q3 - zv.w;
    red[t] = d0 * d0 + d1 * d1 + d2 * d2 + d3 * d3;
    __syncthreads();
    for (int off = 128; off; off >>= 1) {
        if (t < off) red[t] += red[t + off];
        __syncthreads();
    }
    if (t == 0) partials[blockIdx.x] = red[0];
}

// One-hot encodings writeout: one block per row, float2 stores (region is 8B aligned)
__global__ __launch_bounds__(256) void vq_encodings(
    const int* __restrict__ idx_ws, float* __restrict__ enc)
{
    const int row = blockIdx.x;
    const int k = idx_ws[row];
    float* rowp = enc + (size_t)row * KCODES;
    const int col0 = threadIdx.x * 16;
#pragma unroll
    for (int g = 0; g < 8; ++g) {
        int c = col0 + g * 2;
        float2 v; v.x = (k == c) ? 1.0f : 0.0f; v.y = (k == c + 1) ? 1.0f : 0.0f;
        *(float2*)(rowp + c) = v;
    }
}

__global__ __launch_bounds__(256) void vq_finalize(
    const float* __restrict__ partials, const unsigned int* __restrict__ counts,
    float* __restrict__ out)
{
    __shared__ float red[256];
    const int t = threadIdx.x;
    // loss = (1 + 0.25) * mean((quant - x)^2)
    float s = 0.f;
    for (int i = 0; i < 32; ++i) s += partials[t + i * 256];
    red[t] = s;
    __syncthreads();
    for (int off = 128; off; off >>= 1) {
        if (t < off) red[t] += red[t + off];
        __syncthreads();
    }
    if (t == 0) out[OFF_LOSS] = 1.25f * red[0] / 8388608.0f;
    __syncthreads();
    // perplexity = exp(-sum p*log(p+1e-10)), p = count/N
    float h = 0.f;
    for (int i = 0; i < 16; ++i) {
        float p = (float)counts[t + i * 256] * (1.0f / 32768.0f);
        h += p * logf(p + 1e-10f);
    }
    red[t] = h;
    __syncthreads();
    for (int off = 128; off; off >>= 1) {
        if (t < off) red[t] += red[t + off];
        __syncthreads();
    }
    if (t == 0) out[OFF_PERP] = expf(-red[0]);
}

extern "C" void kernel_launch(void* const* d_in, const int* in_sizes, int n_in,
                              void* d_out, int out_size, void* d_ws, size_t ws_size,
                              hipStream_t stream) {
    const float* z  = (const float*)d_in[0];   // [8,256,64,64]
    const float* cb = (const float*)d_in[1];   // [4096,256]
    float* out = (float*)d_out;
    char* ws = (char*)d_ws;

    int* idx_ws          = (int*)(ws + WS_IDX);
    unsigned int* counts = (unsigned int*)(ws + WS_COUNTS);
    float* partials      = (float*)(ws + WS_PARTIALS);
    float* norms         = (float*)(ws + WS_NORMS);

    vq_zero_counts<<<(KCODES + 255) / 256, 256, 0, stream>>>(counts);
    vq_cb_norms<<<KCODES / 128, 128, 0, stream>>>(cb, norms);

    // LDS: A (128 x 260) + 2 x B (32 x 260) = 199,680 bytes (< 320 KB/WGP)
    const size_t shmem = (size_t)(ROWS_PER_WG + 2 * BCODES) * APAD * sizeof(float);
    vq_argmin_kernel<<<N_TOK / ROWS_PER_WG, 128, shmem, stream>>>(
        z, cb, norms, idx_ws, counts, out + OFF_IDX);

    vq_quantize_loss<<<8192, 256, 0, stream>>>(z, cb, idx_ws, out + OFF_QUANT, partials);
    vq_encodings<<<N_TOK, 256, 0, stream>>>(idx_ws, out + OFF_ENC);
    vq_finalize<<<1, 256, 0, stream>>>(partials, counts, out);
}